// fourLoss_31817117728948
// MI455X (gfx1250) — compile-verified
//
#include <hip/hip_runtime.h>

#define NC 52
#define ROWLEN 104
#define NBLOCKS 2048
#define TPB 256
#define WAVES_PER_BLOCK (TPB / 32)

typedef __attribute__((ext_vector_type(2))) float v2f;
typedef __attribute__((ext_vector_type(2))) int   v2i;
typedef __attribute__((ext_vector_type(8))) float v8f;

// Per-wave: tiles of 16 rows. For each 4-wide K chunk:
//   A (16x4 f32) = t tile (exact int->f32), B (4x16 f32) = y tile "transposed".
// ISA layouts for 32-bit A(16x4) and B(4x16) are the same lane mapping:
//   lane L<16  -> (row L,    k+0, k+1)
//   lane L>=16 -> (row L-16, k+2, k+3)
// so both operands are plain 8-byte loads per lane. Diagonal of D gives the
// per-row dot products; a second accumulator uses the half-swapped y columns.
__global__ __launch_bounds__(TPB) void pair_loss_main(
    const float* __restrict__ y, const int* __restrict__ t,
    float* __restrict__ partials, int B)
{
    const int lane        = threadIdx.x & 31;
    const int waveInBlock = threadIdx.x >> 5;
    const int globalWave  = blockIdx.x * WAVES_PER_BLOCK + waveInBlock;
    const int totalWaves  = gridDim.x * WAVES_PER_BLOCK;
    const int numTiles    = (B + 15) >> 4;

    const int m    = lane & 15;      // row within tile
    const int half = lane >> 4;      // which K pair this lane supplies
    // which accumulator component holds this lane's diagonal element
    const int sel  = (lane < 8) ? lane : ((lane >= 24) ? (lane - 24) : -1);

    float s1 = 0.f, s2 = 0.f;

    for (int tile = globalWave; tile < numTiles; tile += totalWaves) {
        const int  row   = tile * 16 + m;
        const bool valid = row < B;
        const size_t base = (size_t)(valid ? row : 0) * ROWLEN;

        v8f c1 = {}; v8f c2 = {};
        #pragma unroll
        for (int kc = 0; kc < 26; ++kc) {
            const int kbase = kc * 4;
            const int k     = kbase + half * 2;
            const int kswap = (kbase < NC) ? (k + NC) : (k - NC);

            v2i tv = *(const v2i*)(t + base + (size_t)k);
            v2f a;
            a.x = valid ? (float)tv.x : 0.f;
            a.y = valid ? (float)tv.y : 0.f;
            v2f b1 = *(const v2f*)(y + base + (size_t)k);
            v2f b2 = *(const v2f*)(y + base + (size_t)kswap);

            c1 = __builtin_amdgcn_wmma_f32_16x16x4_f32(false, a, false, b1,
                                                       (short)0, c1, false, false);
            c2 = __builtin_amdgcn_wmma_f32_16x16x4_f32(false, a, false, b2,
                                                       (short)0, c2, false, false);
        }
        #pragma unroll
        for (int v = 0; v < 8; ++v) {
            s1 += (v == sel) ? c1[v] : 0.f;
            s2 += (v == sel) ? c2[v] : 0.f;
        }
    }

    // wave32 reduction
    for (int off = 16; off > 0; off >>= 1) {
        s1 += __shfl_xor(s1, off, 32);
        s2 += __shfl_xor(s2, off, 32);
    }

    __shared__ float ls1[WAVES_PER_BLOCK], ls2[WAVES_PER_BLOCK];
    if (lane == 0) { ls1[waveInBlock] = s1; ls2[waveInBlock] = s2; }
    __syncthreads();
    if (threadIdx.x == 0) {
        float b1 = 0.f, b2 = 0.f;
        #pragma unroll
        for (int w = 0; w < WAVES_PER_BLOCK; ++w) { b1 += ls1[w]; b2 += ls2[w]; }
        partials[2 * blockIdx.x + 0] = b1;
        partials[2 * blockIdx.x + 1] = b2;
    }
}

__global__ __launch_bounds__(TPB) void pair_loss_final(
    const float* __restrict__ partials, float* __restrict__ out, int nb)
{
    float s1 = 0.f, s2 = 0.f;
    for (int i = threadIdx.x; i < nb; i += TPB) {
        s1 += partials[2 * i + 0];
        s2 += partials[2 * i + 1];
    }
    for (int off = 16; off > 0; off >>= 1) {
        s1 += __shfl_xor(s1, off, 32);
        s2 += __shfl_xor(s2, off, 32);
    }
    __shared__ float ls1[WAVES_PER_BLOCK], ls2[WAVES_PER_BLOCK];
    const int lane = threadIdx.x & 31, w = threadIdx.x >> 5;
    if (lane == 0) { ls1[w] = s1; ls2[w] = s2; }
    __syncthreads();
    if (threadIdx.x == 0) {
        float a = 0.f, b = 0.f;
        #pragma unroll
        for (int i = 0; i < WAVES_PER_BLOCK; ++i) { a += ls1[i]; b += ls2[i]; }
        out[0] = fminf(-0.5f * a, -0.5f * b);
    }
}

extern "C" void kernel_launch(void* const* d_in, const int* in_sizes, int n_in,
                              void* d_out, int out_size, void* d_ws, size_t ws_size,
                              hipStream_t stream) {
    const float* y = (const float*)d_in[0];
    const int*   t = (const int*)d_in[1];
    float*       out = (float*)d_out;
    float*       partials = (float*)d_ws;   // NBLOCKS * 2 floats (16 KB)

    const int B = in_sizes[0] / ROWLEN;     // 1,000,000

    pair_loss_main<<<NBLOCKS, TPB, 0, stream>>>(y, t, partials, B);
    pair_loss_final<<<1, TPB, 0, stream>>>(partials, out, NBLOCKS);
}